// NDStackRNNCore_85315230368216
// MI455X (gfx1250) — compile-verified
//
#include <hip/hip_runtime.h>
#include <hip/hip_bf16.h>
#include <math.h>

typedef float v2f __attribute__((ext_vector_type(2)));
typedef float v4f __attribute__((ext_vector_type(4)));
typedef float v8f __attribute__((ext_vector_type(8)));

#define DIN  188
#define H_   192
#define NACT 576

// ---------------------------------------------------------------------------
// Kernel 0: pack xin = [x | top_stack] into contiguous 64 x 192 rows so the
// WMMA GEMM operand fetch is branch-free and b64-vectorizable.
// ---------------------------------------------------------------------------
__global__ __launch_bounds__(64) void pack_xin(const float* __restrict__ x,
    const float* __restrict__ top, float* __restrict__ xin)
{
  const int m = blockIdx.x;
  for (int k = threadIdx.x; k < 192; k += 64)
    xin[m * 192 + k] = (k < DIN) ? x[m * DIN + k] : top[m * 4 + (k - DIN)];
}

// ---------------------------------------------------------------------------
// Kernel 1: h_new = tanh(xin @ W_ih^T + h @ W_hh^T + b_ih + b_hh)
// M=64, N=192, K=192+192.  One wave per 16x16 tile, v_wmma_f32_16x16x4_f32.
// A (16x4 f32): lanes 0-15 rows M; v0/v1 = K0/K1 (half 0), K2/K3 (half 1)
// B (4x16 f32): v0 = row K0 (lanes 0-15) / K2 (lanes 16-31), v1 = K1/K3
// C/D (16x16):  VGPR r: M=r (lanes 0-15), M=r+8 (lanes 16-31), N = lane%16
// ---------------------------------------------------------------------------
__global__ __launch_bounds__(32) void rnn_gemm(const float* __restrict__ xin,
    const float* __restrict__ h,
    const float* __restrict__ Wih, const float* __restrict__ bih,
    const float* __restrict__ Whh, const float* __restrict__ bhh,
    float* __restrict__ hnew)
{
  const int tile = blockIdx.x;            // 48 tiles: 4 (M) x 12 (N)
  const int tm = (tile & 3) * 16;
  const int tn = (tile >> 2) * 16;
  const int lane = threadIdx.x;
  const int koff = (lane >> 4) * 2;       // K sub-offset per half-wave
  const int l16  = lane & 15;
  const int m = tm + l16;
  const int n = tn + l16;

  const float* A0 = xin + m * H_ + koff;
  const float* B0 = Wih + n * H_ + koff;
  const float* A1 = h   + m * H_ + koff;
  const float* B1 = Whh + n * H_ + koff;

  v8f c = {};
#pragma unroll 4
  for (int kb = 0; kb < 192; kb += 4) {
    v2f a = *(const v2f*)(A0 + kb);
    v2f b = *(const v2f*)(B0 + kb);
    c = __builtin_amdgcn_wmma_f32_16x16x4_f32(false, a, false, b, (short)0, c,
                                              false, false);
  }
#pragma unroll 4
  for (int kb = 0; kb < 192; kb += 4) {
    v2f a = *(const v2f*)(A1 + kb);
    v2f b = *(const v2f*)(B1 + kb);
    c = __builtin_amdgcn_wmma_f32_16x16x4_f32(false, a, false, b, (short)0, c,
                                              false, false);
  }
  const float bias = bih[n] + bhh[n];
  const int half8 = (lane >> 4) * 8;
#pragma unroll
  for (int r = 0; r < 8; ++r) {
    const int mm = tm + r + half8;
    hnew[mm * H_ + n] = tanhf(c[r] + bias);
  }
}

// ---------------------------------------------------------------------------
// Kernel 2: logits = h_new @ W_lin^T + b_lin   (M=64, N=576, K=192)
// ---------------------------------------------------------------------------
__global__ __launch_bounds__(32) void lin_gemm(const float* __restrict__ hnew,
    const float* __restrict__ Wlin, const float* __restrict__ blin,
    float* __restrict__ logits)
{
  const int tile = blockIdx.x;            // 144 tiles: 4 (M) x 36 (N)
  const int tm = (tile & 3) * 16;
  const int tn = (tile >> 2) * 16;
  const int lane = threadIdx.x;
  const int koff = (lane >> 4) * 2;
  const int l16  = lane & 15;
  const int m = tm + l16;
  const int n = tn + l16;

  const float* A = hnew + m * H_ + koff;
  const float* B = Wlin + n * H_ + koff;

  v8f c = {};
#pragma unroll 4
  for (int kb = 0; kb < 192; kb += 4) {
    v2f a = *(const v2f*)(A + kb);
    v2f b = *(const v2f*)(B + kb);
    c = __builtin_amdgcn_wmma_f32_16x16x4_f32(false, a, false, b, (short)0, c,
                                              false, false);
  }
  const float bias = blin[n];
  const int half8 = (lane >> 4) * 8;
#pragma unroll
  for (int r = 0; r < 8; ++r) {
    const int mm = tm + r + half8;
    logits[mm * NACT + n] = c[r] + bias;
  }
}

// ---------------------------------------------------------------------------
// Kernel 3: row-wise softmax over 576 actions, in place. One block per row.
// ---------------------------------------------------------------------------
__global__ __launch_bounds__(256) void softmax_rows(float* __restrict__ logits)
{
  __shared__ float red[256];
  const int b = blockIdx.x, t = threadIdx.x;
  float* row = logits + b * NACT;
  float lv[3];
  float mx = -1e30f;
#pragma unroll
  for (int j = 0; j < 3; ++j) {
    const int idx = t + j * 256;
    lv[j] = (idx < NACT) ? row[idx] : -1e30f;
    mx = fmaxf(mx, lv[j]);
  }
  red[t] = mx; __syncthreads();
  for (int off = 128; off; off >>= 1) {
    if (t < off) red[t] = fmaxf(red[t], red[t + off]);
    __syncthreads();
  }
  mx = red[0]; __syncthreads();
  float ev[3];
  float s = 0.f;
#pragma unroll
  for (int j = 0; j < 3; ++j) {
    const int idx = t + j * 256;
    ev[j] = (idx < NACT) ? __expf(lv[j] - mx) : 0.f;
    s += ev[j];
  }
  red[t] = s; __syncthreads();
  for (int off = 128; off; off >>= 1) {
    if (t < off) red[t] += red[t + off];
    __syncthreads();
  }
  const float inv = 1.f / red[0];
#pragma unroll
  for (int j = 0; j < 3; ++j) {
    const int idx = t + j * 256;
    if (idx < NACT) row[idx] = ev[j] * inv;
  }
}

// ---------------------------------------------------------------------------
// Kernel 4: non-temporal b128 streaming copy (the 256 MB gamma pass-through).
// NT hints keep the stream from evicting the gamma slices the contraction
// kernels re-read out of L2.
// ---------------------------------------------------------------------------
__global__ void copy_f4(const v4f* __restrict__ src, v4f* __restrict__ dst,
                        long n)
{
  long i = (long)blockIdx.x * blockDim.x + threadIdx.x;
  const long stride = (long)gridDim.x * blockDim.x;
  for (; i < n; i += stride) {
    v4f v = __builtin_nontemporal_load(src + i);
    __builtin_nontemporal_store(v, dst + i);
  }
}

// ---------------------------------------------------------------------------
// Kernel 5: tmp[b,k,u,y,r] = sum_{s,z} gamma[b,k,tsm1,u,y,s,z] * pop[b,s,z,r]
// ---------------------------------------------------------------------------
__global__ __launch_bounds__(256) void tmp_kernel(const float* __restrict__ gamma,
    const float* __restrict__ actions, const int* __restrict__ tsptr,
    float* __restrict__ tmp)
{
  const int idx = blockIdx.x * 256 + threadIdx.x;   // 65536 total
  const int tsm1 = tsptr[0];
  const int b   = idx >> 12;
  const int rem = idx & 4095;
  const int k = rem >> 6;
  const int u = (rem >> 4) & 3;
  const int y = (rem >> 2) & 3;
  const int r = rem & 3;
  const float* g  = gamma + (((long)(b * 64 + k) * 64 + tsm1) << 8) + u * 64 + y * 16;
  const float* pp = actions + b * NACT + 256;       // pop block
  float acc = 0.f;
#pragma unroll
  for (int s = 0; s < 4; ++s)
#pragma unroll
    for (int z = 0; z < 4; ++z)
      acc += g[s * 4 + z] * pp[s * 16 + z * 4 + r];
  tmp[idx] = acc;
}

// ---------------------------------------------------------------------------
// Kernel 6: gamma_t[b,i,q,x,r,y] = push + pop + repl, written into
// out_gamma[b, i, ts, ...].  One block per (b,i); gcol & repl staged in LDS.
// ---------------------------------------------------------------------------
__global__ __launch_bounds__(256) void gamma_t_kernel(const float* __restrict__ gamma,
    const float* __restrict__ actions, const float* __restrict__ tmp,
    const int* __restrict__ tsptr, float* __restrict__ outg)
{
  __shared__ float gcol_s[256];
  __shared__ float repl_s[256];
  const int bi = blockIdx.x;                 // b*64 + i
  const int b = bi >> 6, i = bi & 63;
  const int t = threadIdx.x;
  const int tsm1 = tsptr[0];
  const long base_bi = (long)bi * 64;        // row index space (each row = 256)
  gcol_s[t] = gamma[(base_bi + tsm1) * 256 + t];
  repl_s[t] = actions[b * NACT + 320 + t];
  __syncthreads();
  const int q = t >> 6, xx = (t >> 4) & 3, r = (t >> 2) & 3, y = t & 3;
  float val = 0.f;
  if (i == tsm1) val += actions[b * NACT + q * 64 + xx * 16 + r * 4 + y]; // push_t
  // repl_t
#pragma unroll
  for (int s = 0; s < 4; ++s)
#pragma unroll
    for (int z = 0; z < 4; ++z)
      val += gcol_s[q * 64 + xx * 16 + s * 4 + z] * repl_s[s * 64 + z * 16 + r * 4 + y];
  // pop_t: k in (i, tsm1)
  const float* tb = tmp + b * 4096;
  for (int k = i + 1; k < tsm1; ++k) {
    const float* g  = gamma + (base_bi + k) * 256 + q * 64 + xx * 16 + y;
    const float* tk = tb + k * 64 + y * 4 + r;
    float acc = 0.f;
#pragma unroll
    for (int u = 0; u < 4; ++u) acc += g[u * 4] * tk[u * 16];
    val += acc;
  }
  outg[(base_bi + tsm1 + 1) * 256 + t] = val;
}

// ---------------------------------------------------------------------------
// Kernel 7: alpha_t[b,r,y] = sum_{i,q,x} alpha[b,i,q,x] * gamma_t[b,i,q,x,r,y]
// then new_alpha[:,ts] and normalized obs.  One block per b.
// ---------------------------------------------------------------------------
__global__ __launch_bounds__(256) void alpha_obs_kernel(const float* __restrict__ alpha,
    const float* __restrict__ outg, const int* __restrict__ tsptr,
    float* __restrict__ new_alpha, float* __restrict__ obs)
{
  __shared__ float red[256];
  __shared__ float ob[4];
  const int b = blockIdx.x, t = threadIdx.x;
  const int ts = tsptr[0] + 1;
  const int ry = t & 15, ii = t >> 4;
  float p = 0.f;
  for (int i = ii; i < 64; i += 16) {
    const float* g = outg + ((long)(b * 64 + i) * 64 + ts) * 256 + ry;
    const float* a = alpha + b * 1024 + i * 16;
#pragma unroll
    for (int qx = 0; qx < 16; ++qx) p += a[qx] * g[qx * 16];
  }
  red[t] = p; __syncthreads();
  for (int off = 128; off >= 16; off >>= 1) {
    if (t < off) red[t] += red[t + off];
    __syncthreads();
  }
  if (t < 16) new_alpha[b * 1024 + ts * 16 + t] = red[t];
  if (t < 4)  ob[t] = red[t] + red[4 + t] + red[8 + t] + red[12 + t];
  __syncthreads();
  if (t < 4) {
    const float denom = ob[0] + ob[1] + ob[2] + ob[3] + 1e-5f;
    obs[b * 4 + t] = ob[t] / denom;
  }
}

// ---------------------------------------------------------------------------
extern "C" void kernel_launch(void* const* d_in, const int* in_sizes, int n_in,
                              void* d_out, int out_size, void* d_ws, size_t ws_size,
                              hipStream_t stream)
{
  const float* x     = (const float*)d_in[0];
  const float* h     = (const float*)d_in[1];
  const float* gamma = (const float*)d_in[2];
  const float* alpha = (const float*)d_in[3];
  const float* top   = (const float*)d_in[4];
  const float* Wih   = (const float*)d_in[5];
  const float* bih   = (const float*)d_in[6];
  const float* Whh   = (const float*)d_in[7];
  const float* bhh   = (const float*)d_in[8];
  const float* Wlin  = (const float*)d_in[9];
  const float* blin  = (const float*)d_in[10];
  const int*   tsp   = (const int*)d_in[11];

  float* out  = (float*)d_out;
  float* hnew = out;                                   // 64*192      = 12288
  float* outg = out + 12288;                           // 64*64*64*256= 67108864
  float* outa = out + 12288 + 67108864L;               // 64*64*16    = 65536
  float* outo = out + 12288 + 67108864L + 65536;       // 64*4        = 256

  float* logits = (float*)d_ws;                        // 64*576 = 36864 floats
  float* tmpb   = logits + 36864;                      // 65536 floats
  float* xin    = tmpb + 65536;                        // 64*192 = 12288 floats

  // Dense RNN + linear head on the WMMA pipe (fp32, exact precision)
  pack_xin<<<64, 64, 0, stream>>>(x, top, xin);
  rnn_gemm<<<48, 32, 0, stream>>>(xin, h, Wih, bih, Whh, bhh, hnew);
  lin_gemm<<<144, 32, 0, stream>>>(hnew, Wlin, blin, logits);
  softmax_rows<<<64, 256, 0, stream>>>(logits);

  // Bandwidth-dominant pass-through copies (non-temporal b128)
  copy_f4<<<4096, 256, 0, stream>>>((const v4f*)gamma, (v4f*)outg,
                                    67108864L / 4);
  copy_f4<<<64, 256, 0, stream>>>((const v4f*)alpha, (v4f*)outa,
                                  65536L / 4);

  // Stack-transition tensor contractions
  tmp_kernel<<<256, 256, 0, stream>>>(gamma, logits, tsp, tmpb);
  gamma_t_kernel<<<4096, 256, 0, stream>>>(gamma, logits, tmpb, tsp, outg);
  alpha_obs_kernel<<<64, 256, 0, stream>>>(alpha, outg, tsp, outa, outo);
}